// GraphAttentionEncoder_32220844655112
// MI455X (gfx1250) — compile-verified
//
#include <hip/hip_runtime.h>
#include <hip/hip_bf16.h>

// ---------------------------------------------------------------------------
// GAT encoder for MI455X (gfx1250, wave32).
// Dense GEMMs use V_WMMA_F32_16X16X4_F32 (f32 tensor path, matches reference
// precision). The per-block A tile (16 x K, shared by all 8 waves of a block)
// is staged into LDS via the Tensor Data Mover (tensor_load_to_lds +
// s_wait_tensorcnt), removing 8x-redundant cache reads. Attention-score GEMMs
// are algebraically folded away:
//   score[e,h] = (h@Wn@a_s)[src] + (h@Wn@a_t)[tgt] + (e_enc@(We@a_e))[e]
// so ep (320000x256 per layer) is never materialized; e_enc is fused into a
// wave-per-edge kernel emitting only 16 floats/edge.
// ---------------------------------------------------------------------------

#define N_NODES   20000
#define N_EDGES   320000
#define HID       256
#define EHID      128
#define NODE_DIM  64
#define EDGE_DIM  32
#define NLAYERS   4
#define NHEADS    4
#define HD        64

typedef __attribute__((ext_vector_type(2))) float v2f;
typedef __attribute__((ext_vector_type(8))) float v8f;
typedef __attribute__((ext_vector_type(4))) unsigned int u32x4;
typedef __attribute__((ext_vector_type(4))) int i32x4;
typedef __attribute__((ext_vector_type(8))) int i32x8;

#if defined(__has_builtin)
#if __has_builtin(__builtin_amdgcn_tensor_load_to_lds)
#define HAVE_TDM 1
#endif
#endif

__device__ __forceinline__ void wait_tensorcnt0() {
#if defined(__has_builtin) && __has_builtin(__builtin_amdgcn_s_wait_tensorcnt)
  __builtin_amdgcn_s_wait_tensorcnt(0);
#else
  asm volatile("s_wait_tensorcnt 0x0" ::: "memory");
#endif
}

__device__ __forceinline__ float wave_red_sum(float v) {
#pragma unroll
  for (int off = 16; off > 0; off >>= 1) v += __shfl_xor(v, off, 32);
  return v;
}

// ---------------------------------------------------------------------------
// Generic f32 WMMA GEMM: C[M,N] = A[M,K] @ B[K,N] (+ bias[N] if non-null).
// One wave computes a 16x16 tile via V_WMMA_F32_16X16X4_F32.
// Block = 256 threads = 8 waves, tiles 16 rows x 128 cols. The shared 16xK
// A tile is DMA'd into LDS by the TDM (one descriptor, issued by wave 0).
// A frag (16x4): lanes 0-15 rows M=lane (K=k0,k0+1); lanes 16-31 (K=k0+2,+3)
// B frag (4x16): symmetric; C/D per ISA 7.12.2 (8 VGPR, M=v / v+8 split).
// Grid: x = N/128, y = M/16. M,N,K multiples of 16/16/4.
// ---------------------------------------------------------------------------
__global__ void gemm_wmma_f32(const float* __restrict__ A,
                              const float* __restrict__ B,
                              const float* __restrict__ bias,
                              float* __restrict__ C,
                              int M, int N, int K) {
  __shared__ float As[16 * 256];  // max K = 256 -> 16 KB

  const float* Ablk = A + (size_t)blockIdx.y * 16 * K;

#if HAVE_TDM
  if (threadIdx.x < 32) {  // wave 0 issues one TDM descriptor for the block
    const unsigned long long ga = (unsigned long long)(uintptr_t)Ablk;
    const unsigned int ldsoff = (unsigned int)(uintptr_t)(&As[0]);
    u32x4 g0;
    g0.x = 1u;                                   // count=1, user D#
    g0.y = ldsoff;                               // lds_addr (bytes)
    g0.z = (unsigned int)ga;                     // global_addr[31:0]
    g0.w = (unsigned int)((ga >> 32) & 0x01FFFFFFu) | 0x80000000u; // addr[56:32] | type=2
    i32x8 g1;
    g1[0] = (int)(2u << 16);                     // wg_mask=0, data_size=4B
    g1[1] = (int)(((unsigned)K & 0xFFFFu) << 16);        // tensor_dim0 lo16 @[63:48]
    g1[2] = (int)((((unsigned)K >> 16) & 0xFFFFu) | (16u << 16)); // dim0 hi | tensor_dim1=16 lo
    g1[3] = (int)(((unsigned)K & 0xFFFFu) << 16);        // tensor_dim1 hi=0 | tile_dim0=K
    g1[4] = 16;                                  // tile_dim1=16, tile_dim2=0
    g1[5] = K;                                   // tensor_dim0_stride lo32 = K
    g1[6] = 0;                                   // stride0 hi | stride1 lo (unused, 2D)
    g1[7] = 0;
    i32x4 z4 = {0, 0, 0, 0};
#if __clang_major__ >= 23
    i32x8 z8 = {0, 0, 0, 0, 0, 0, 0, 0};
    __builtin_amdgcn_tensor_load_to_lds(g0, g1, z4, z4, z8, 0);
#else
    __builtin_amdgcn_tensor_load_to_lds(g0, g1, z4, z4, 0);
#endif
    wait_tensorcnt0();
  }
  __syncthreads();
#else
  for (int i = threadIdx.x; i < 16 * K; i += blockDim.x) As[i] = Ablk[i];
  __syncthreads();
#endif

  const int lane  = threadIdx.x & 31;
  const int wave  = threadIdx.x >> 5;
  const int tileN = blockIdx.x * 128 + wave * 16;
  const int row   = lane & 15;            // M-row within A frag / col within B,C
  const int koff  = (lane >> 4) << 1;     // 0 for lanes 0-15, 2 for lanes 16-31

  const float* Ar = As + row * K;
  const float* Bp = B + (size_t)(tileN + row);

  v8f acc = {};
#pragma unroll 4
  for (int k0 = 0; k0 < K; k0 += 4) {
    v2f a, b;
    a.x = Ar[k0 + koff];                      // ds_load_b64 from staged tile
    a.y = Ar[k0 + koff + 1];
    b.x = Bp[(size_t)(k0 + koff) * N];
    b.y = Bp[(size_t)(k0 + koff + 1) * N];
    acc = __builtin_amdgcn_wmma_f32_16x16x4_f32(false, a, false, b,
                                                (short)0, acc, false, false);
  }

  const int mBase = blockIdx.y * 16 + ((lane >> 4) << 3);
  float* Cp = C + (size_t)mBase * N + tileN + row;
  const float bv = bias ? bias[tileN + row] : 0.0f;
#pragma unroll
  for (int v = 0; v < 8; ++v) Cp[(size_t)v * N] = acc[v] + bv;
}

// ---------------------------------------------------------------------------
// Row LayerNorm + ReLU in place. One wave per row, D in {128,256}.
// ---------------------------------------------------------------------------
__global__ void ln_relu_rows(float* __restrict__ x, const float* __restrict__ g,
                             const float* __restrict__ b, int rows, int D) {
  const int r    = (blockIdx.x * blockDim.x + threadIdx.x) >> 5;
  const int lane = threadIdx.x & 31;
  if (r >= rows) return;
  float* xp = x + (size_t)r * D;
  const int per = D >> 5;
  float sum = 0.f, sq = 0.f;
  for (int i = 0; i < per; ++i) {
    float v = xp[i * 32 + lane];
    sum += v; sq += v * v;
  }
  sum = wave_red_sum(sum);
  sq  = wave_red_sum(sq);
  const float mu   = sum / (float)D;
  const float var  = sq / (float)D - mu * mu;
  const float rstd = rsqrtf(var + 1e-5f);
  for (int i = 0; i < per; ++i) {
    int idx = i * 32 + lane;
    float v = (xp[idx] - mu) * rstd * g[idx] + b[idx];
    xp[idx] = fmaxf(v, 0.0f);
  }
}

// ---------------------------------------------------------------------------
// Fold We with a_e for all layers: Vall[l][k][h] = sum_d We[l][k][h*64+d]*a_e[l][h][d]
// ---------------------------------------------------------------------------
__global__ void fold_V_all(const float* __restrict__ We,
                           const float* __restrict__ att_a,
                           float* __restrict__ Vall) {
  int idx = blockIdx.x * blockDim.x + threadIdx.x;
  if (idx >= NLAYERS * EHID * NHEADS) return;
  int l = idx >> 9, k = (idx >> 2) & (EHID - 1), h = idx & 3;
  const float* w = We + (size_t)l * EHID * HID + (size_t)k * HID + h * HD;
  const float* a = att_a + (size_t)l * NHEADS * 192 + h * 192 + 128;
  float s = 0.f;
#pragma unroll 8
  for (int d = 0; d < HD; ++d) s += w[d] * a[d];
  Vall[idx] = s;
}

// ---------------------------------------------------------------------------
// Fused edge encoder: per edge (one wave): e = relu(LN(ef@W + b)); then
// sE[e][l][h] = dot(e, Vall[l][:,h]). Output 16 floats/edge; e_enc never stored.
// ---------------------------------------------------------------------------
__global__ void edge_encode_fold(const float* __restrict__ ef,
                                 const float* __restrict__ eenc_w,
                                 const float* __restrict__ eenc_b,
                                 const float* __restrict__ eenc_g,
                                 const float* __restrict__ eenc_beta,
                                 const float4* __restrict__ Vall4,
                                 float* __restrict__ sE, int Ecnt) {
  __shared__ float  W[EDGE_DIM * EHID];
  __shared__ float  bb[EHID], gg[EHID], be[EHID];
  __shared__ float4 Vs[NLAYERS * EHID];
  for (int i = threadIdx.x; i < EDGE_DIM * EHID; i += blockDim.x) W[i] = eenc_w[i];
  for (int i = threadIdx.x; i < EHID; i += blockDim.x) {
    bb[i] = eenc_b[i]; gg[i] = eenc_g[i]; be[i] = eenc_beta[i];
  }
  for (int i = threadIdx.x; i < NLAYERS * EHID; i += blockDim.x) Vs[i] = Vall4[i];
  __syncthreads();

  const int e    = (blockIdx.x * blockDim.x + threadIdx.x) >> 5;
  const int lane = threadIdx.x & 31;
  if (e >= Ecnt) return;

  const float xin = ef[(size_t)e * EDGE_DIM + lane];
  const int c0 = lane, c1 = lane + 32, c2 = lane + 64, c3 = lane + 96;
  float o0 = bb[c0], o1 = bb[c1], o2 = bb[c2], o3 = bb[c3];
#pragma unroll
  for (int k = 0; k < EDGE_DIM; ++k) {
    float xv = __shfl(xin, k, 32);
    const float* wr = W + k * EHID;
    o0 += xv * wr[c0]; o1 += xv * wr[c1]; o2 += xv * wr[c2]; o3 += xv * wr[c3];
  }
  float sum = wave_red_sum(o0 + o1 + o2 + o3);
  float sq  = wave_red_sum(o0 * o0 + o1 * o1 + o2 * o2 + o3 * o3);
  const float mu   = sum * (1.0f / EHID);
  const float var  = sq * (1.0f / EHID) - mu * mu;
  const float rstd = rsqrtf(var + 1e-5f);
  o0 = fmaxf((o0 - mu) * rstd * gg[c0] + be[c0], 0.f);
  o1 = fmaxf((o1 - mu) * rstd * gg[c1] + be[c1], 0.f);
  o2 = fmaxf((o2 - mu) * rstd * gg[c2] + be[c2], 0.f);
  o3 = fmaxf((o3 - mu) * rstd * gg[c3] + be[c3], 0.f);

  float4 p[NLAYERS];
#pragma unroll
  for (int l = 0; l < NLAYERS; ++l) {
    const float4* vl = Vs + l * EHID;
    float4 v0 = vl[c0], v1 = vl[c1], v2 = vl[c2], v3 = vl[c3];
    p[l].x = o0 * v0.x + o1 * v1.x + o2 * v2.x + o3 * v3.x;
    p[l].y = o0 * v0.y + o1 * v1.y + o2 * v2.y + o3 * v3.y;
    p[l].z = o0 * v0.z + o1 * v1.z + o2 * v2.z + o3 * v3.z;
    p[l].w = o0 * v0.w + o1 * v1.w + o2 * v2.w + o3 * v3.w;
  }
#pragma unroll
  for (int l = 0; l < NLAYERS; ++l) {
    p[l].x = wave_red_sum(p[l].x);
    p[l].y = wave_red_sum(p[l].y);
    p[l].z = wave_red_sum(p[l].z);
    p[l].w = wave_red_sum(p[l].w);
  }
  if (lane == 0) {
    float4* o = (float4*)(sE + (size_t)e * 16);
#pragma unroll
    for (int l = 0; l < NLAYERS; ++l) o[l] = p[l];
  }
}

// ---------------------------------------------------------------------------
// Per-node attention scalars: s_s[n,h] = hp[n,h,:].a_s[h], s_t likewise.
// One wave per node.
// ---------------------------------------------------------------------------
__global__ void node_scores(const float* __restrict__ hp,
                            const float* __restrict__ att, // layer slice, H x 192
                            float* __restrict__ s_s, float* __restrict__ s_t,
                            int rows) {
  __shared__ float as[HID], at[HID];
  for (int i = threadIdx.x; i < HID; i += blockDim.x) {
    int h = i >> 6, d = i & 63;
    as[i] = att[h * 192 + d];
    at[i] = att[h * 192 + 64 + d];
  }
  __syncthreads();
  const int n    = (blockIdx.x * blockDim.x + threadIdx.x) >> 5;
  const int lane = threadIdx.x & 31;
  if (n >= rows) return;
  const float4* hv = (const float4*)(hp + (size_t)n * HID);
  float4 x0 = hv[lane * 2], x1 = hv[lane * 2 + 1];
  const int base = lane * 8;
  float ps = x0.x * as[base]   + x0.y * as[base+1] + x0.z * as[base+2] + x0.w * as[base+3]
           + x1.x * as[base+4] + x1.y * as[base+5] + x1.z * as[base+6] + x1.w * as[base+7];
  float pt = x0.x * at[base]   + x0.y * at[base+1] + x0.z * at[base+2] + x0.w * at[base+3]
           + x1.x * at[base+4] + x1.y * at[base+5] + x1.z * at[base+6] + x1.w * at[base+7];
#pragma unroll
  for (int off = 4; off > 0; off >>= 1) {  // reduce within 8-lane head group
    ps += __shfl_xor(ps, off, 32);
    pt += __shfl_xor(pt, off, 32);
  }
  if ((lane & 7) == 0) {
    s_s[n * NHEADS + (lane >> 3)] = ps;
    s_t[n * NHEADS + (lane >> 3)] = pt;
  }
}

// ---------------------------------------------------------------------------
// Edge scores: leaky(s_s[src]+s_t[tgt]+sE), store; atomic segment-max into m
// (safe int-punned max: m init 0, only positive scores pushed); degree count.
// ---------------------------------------------------------------------------
__global__ void edge_scores(const float* __restrict__ sE, int layer,
                            const float* __restrict__ s_s,
                            const float* __restrict__ s_t,
                            const int* __restrict__ srcIdx,
                            const int* __restrict__ tgtIdx,
                            float* __restrict__ score, float* __restrict__ m,
                            float* __restrict__ deg, int Ecnt) {
  int e = blockIdx.x * blockDim.x + threadIdx.x;
  if (e >= Ecnt) return;
  const int s = srcIdx[e], t = tgtIdx[e];
  float4 se = ((const float4*)sE)[(size_t)e * NLAYERS + layer];
  float4 ss = ((const float4*)s_s)[s];
  float4 st = ((const float4*)s_t)[t];
  float sc[4] = { ss.x + st.x + se.x, ss.y + st.y + se.y,
                  ss.z + st.z + se.z, ss.w + st.w + se.w };
#pragma unroll
  for (int h = 0; h < NHEADS; ++h) {
    float v = sc[h];
    v = (v >= 0.f) ? v : 0.2f * v;
    sc[h] = v;
    if (v > 0.f) atomicMax((int*)m + (size_t)t * NHEADS + h, __float_as_int(v));
  }
  ((float4*)score)[e] = make_float4(sc[0], sc[1], sc[2], sc[3]);
  atomicAdd(deg + t, 1.0f);
}

__global__ void exp_sum(const float* __restrict__ score,
                        const int* __restrict__ tgtIdx,
                        const float* __restrict__ m,
                        float* __restrict__ denom, int Ecnt) {
  int e = blockIdx.x * blockDim.x + threadIdx.x;
  if (e >= Ecnt) return;
  const int t = tgtIdx[e];
  float4 sc = ((const float4*)score)[e];
  float4 mm = ((const float4*)m)[t];
  float* d = denom + (size_t)t * NHEADS;
  atomicAdd(d + 0, __expf(sc.x - mm.x));
  atomicAdd(d + 1, __expf(sc.y - mm.y));
  atomicAdd(d + 2, __expf(sc.z - mm.z));
  atomicAdd(d + 3, __expf(sc.w - mm.w));
}

__global__ void denom_finish(const float* __restrict__ deg,
                             const float* __restrict__ m,
                             float* __restrict__ denom, float Ef, int NH) {
  int i = blockIdx.x * blockDim.x + threadIdx.x;
  if (i >= NH) return;
  denom[i] += (Ef - deg[i >> 2]) * __expf(-m[i]);
}

// ---------------------------------------------------------------------------
// Aggregation: h_next[tgt] += w * hp[src]; one wave per edge (8 f32/lane).
// hp/h_next are L2-resident (20.5 MB each); atomics stay on-chip.
// ---------------------------------------------------------------------------
__global__ void aggregate(const float* __restrict__ hp,
                          const float* __restrict__ score,
                          const float* __restrict__ m,
                          const float* __restrict__ denom,
                          const int* __restrict__ srcIdx,
                          const int* __restrict__ tgtIdx,
                          float* __restrict__ hnext, int Ecnt) {
  const int e    = (blockIdx.x * blockDim.x + threadIdx.x) >> 5;
  const int lane = threadIdx.x & 31;
  if (e >= Ecnt) return;
  const int s = srcIdx[e], t = tgtIdx[e];
  const int h = lane >> 3;
  const float w = __expf(score[(size_t)e * NHEADS + h] - m[(size_t)t * NHEADS + h])
                / denom[(size_t)t * NHEADS + h];
  const float4* hs = (const float4*)(hp + (size_t)s * HID) + lane * 2;
  float4 x0 = hs[0], x1 = hs[1];
  float* out = hnext + (size_t)t * HID + lane * 8;
  atomicAdd(out + 0, w * x0.x); atomicAdd(out + 1, w * x0.y);
  atomicAdd(out + 2, w * x0.z); atomicAdd(out + 3, w * x0.w);
  atomicAdd(out + 4, w * x1.x); atomicAdd(out + 5, w * x1.y);
  atomicAdd(out + 6, w * x1.z); atomicAdd(out + 7, w * x1.w);
}

__global__ void pool_sum(const float* __restrict__ h, float* __restrict__ pooled,
                         int rows) {
  const int col = threadIdx.x;
  float s = 0.f;
  for (int r = blockIdx.x; r < rows; r += gridDim.x) s += h[(size_t)r * HID + col];
  atomicAdd(pooled + col, s);
}

__global__ void mlp_head(const float* __restrict__ pooled,
                         const float* __restrict__ w1, const float* __restrict__ b1,
                         const float* __restrict__ w2, const float* __restrict__ b2,
                         float* __restrict__ out, float invN) {
  __shared__ float p[HID];
  __shared__ float t[HID];
  const int j = threadIdx.x;
  p[j] = pooled[j] * invN;
  __syncthreads();
  float s = b1[j];
  for (int k = 0; k < HID; ++k) s += p[k] * w1[k * HID + j];
  t[j] = fmaxf(s, 0.f);
  __syncthreads();
  float o = b2[j];
  for (int k = 0; k < HID; ++k) o += t[k] * w2[k * HID + j];
  out[j] = o;
}

// ---------------------------------------------------------------------------
extern "C" void kernel_launch(void* const* d_in, const int* in_sizes, int n_in,
                              void* d_out, int out_size, void* d_ws, size_t ws_size,
                              hipStream_t stream) {
  const float* node_feats = (const float*)d_in[0];
  const float* edge_feats = (const float*)d_in[1];
  const int*   eidx       = (const int*)d_in[2];
  const float* enc_w      = (const float*)d_in[3];
  const float* enc_b      = (const float*)d_in[4];
  const float* enc_g      = (const float*)d_in[5];
  const float* enc_beta   = (const float*)d_in[6];
  const float* eenc_w     = (const float*)d_in[7];
  const float* eenc_b     = (const float*)d_in[8];
  const float* eenc_g     = (const float*)d_in[9];
  const float* eenc_beta  = (const float*)d_in[10];
  const float* Wn         = (const float*)d_in[11];
  const float* We         = (const float*)d_in[12];
  const float* att_a      = (const float*)d_in[13];
  const float* pool_w1    = (const float*)d_in[14];
  const float* pool_b1    = (const float*)d_in[15];
  const float* pool_w2    = (const float*)d_in[16];
  const float* pool_b2    = (const float*)d_in[17];

  const int* srcIdx = eidx;
  const int* tgtIdx = eidx + N_EDGES;

  // workspace carve-out (~88 MB)
  char* ws = (char*)d_ws;
  auto carve = [&](size_t bytes) -> float* {
    float* p = (float*)ws;
    ws += (bytes + 255) & ~(size_t)255;
    return p;
  };
  float* hp     = carve((size_t)N_NODES * HID * 4);
  float* hA     = carve((size_t)N_NODES * HID * 4);
  float* hB     = carve((size_t)N_NODES * HID * 4);
  float* sE     = carve((size_t)N_EDGES * NLAYERS * NHEADS * 4);
  float* score  = carve((size_t)N_EDGES * NHEADS * 4);
  float* s_s    = carve((size_t)N_NODES * NHEADS * 4);
  float* s_t    = carve((size_t)N_NODES * NHEADS * 4);
  float* mbuf   = carve((size_t)N_NODES * NHEADS * 4);
  float* denom  = carve((size_t)N_NODES * NHEADS * 4);
  float* deg    = carve((size_t)N_NODES * 4);
  float* Vall   = carve((size_t)NLAYERS * EHID * NHEADS * 4);
  float* pooled = carve(HID * 4);

  // --- edge path: fold + fused encoder (e_enc never materialized) ---
  fold_V_all<<<(NLAYERS * EHID * NHEADS + 255) / 256, 256, 0, stream>>>(We, att_a, Vall);
  edge_encode_fold<<<N_EDGES / 8, 256, 0, stream>>>(
      edge_feats, eenc_w, eenc_b, eenc_g, eenc_beta, (const float4*)Vall, sE, N_EDGES);

  // --- node encoder: WMMA GEMM + LN/ReLU ---
  gemm_wmma_f32<<<dim3(HID / 128, N_NODES / 16), 256, 0, stream>>>(
      node_feats, enc_w, enc_b, hA, N_NODES, HID, NODE_DIM);
  ln_relu_rows<<<N_NODES / 8, 256, 0, stream>>>(hA, enc_g, enc_beta, N_NODES, HID);

  float* hc = hA;
  float* hn = hB;
  for (int l = 0; l < NLAYERS; ++l) {
    gemm_wmma_f32<<<dim3(HID / 128, N_NODES / 16), 256, 0, stream>>>(
        hc, Wn + (size_t)l * HID * HID, nullptr, hp, N_NODES, HID, HID);
    node_scores<<<N_NODES / 8, 256, 0, stream>>>(
        hp, att_a + (size_t)l * NHEADS * 192, s_s, s_t, N_NODES);

    hipMemsetAsync(mbuf,  0, (size_t)N_NODES * NHEADS * 4, stream);
    hipMemsetAsync(denom, 0, (size_t)N_NODES * NHEADS * 4, stream);
    hipMemsetAsync(deg,   0, (size_t)N_NODES * 4, stream);
    hipMemsetAsync(hn,    0, (size_t)N_NODES * HID * 4, stream);

    edge_scores<<<(N_EDGES + 255) / 256, 256, 0, stream>>>(
        sE, l, s_s, s_t, srcIdx, tgtIdx, score, mbuf, deg, N_EDGES);
    exp_sum<<<(N_EDGES + 255) / 256, 256, 0, stream>>>(
        score, tgtIdx, mbuf, denom, N_EDGES);
    denom_finish<<<(N_NODES * NHEADS + 255) / 256, 256, 0, stream>>>(
        deg, mbuf, denom, (float)N_EDGES, N_NODES * NHEADS);
    aggregate<<<N_EDGES / 8, 256, 0, stream>>>(
        hp, score, mbuf, denom, srcIdx, tgtIdx, hn, N_EDGES);

    float* tmp = hc; hc = hn; hn = tmp;
  }

  hipMemsetAsync(pooled, 0, HID * 4, stream);
  pool_sum<<<256, 256, 0, stream>>>(hc, pooled, N_NODES);
  mlp_head<<<1, 256, 0, stream>>>(pooled, pool_w1, pool_b1, pool_w2, pool_b2,
                                  (float*)d_out, 1.0f / (float)N_NODES);
}